// WeightedRuleLayer_30605936951443
// MI455X (gfx1250) — compile-verified
//
#include <hip/hip_runtime.h>
#include <hip/hip_bf16.h>
#include <math.h>

// MI455X (gfx1250, wave32) implementation.
//
// Roofline: ~300 MB of traffic @ 23.3 TB/s ~= 13 us; compute (64 MFLOP + tanh)
// is negligible -> pure bandwidth problem. Strategy:
//   Pass 1: materialize the 16 MB concat table in d_ws (fits global 192 MB L2),
//           so pass-2 random gathers are L2 hits instead of double indirection.
//   Pass 2: stream concat_idx with non-temporal b64 loads (don't evict the
//           table from L2), gather from the table, reduce K=8 with two chained
//           V_WMMA_F32_16X16X4_F32 (f32 precision == reference), tanh, NT store.

typedef __attribute__((ext_vector_type(2))) float v2f;
typedef __attribute__((ext_vector_type(4))) float v4f;
typedef __attribute__((ext_vector_type(8))) float v8f;
typedef __attribute__((ext_vector_type(2))) int   v2i;
typedef __attribute__((ext_vector_type(4))) int   v4i;

// ---------------------------------------------------------------------------
// Pass 1: table[i] = (i < src) ? values_a[idx_a[i]] : values_b[idx_b[i-src]]
// 4 elements per thread; idx stream read non-temporally (read once).
// ---------------------------------------------------------------------------
__global__ void gather_concat_kernel(const float* __restrict__ va,
                                     const float* __restrict__ vb,
                                     const int*   __restrict__ ia,
                                     const int*   __restrict__ ib,
                                     float*       __restrict__ table,
                                     int src_size, int total4) {
    int t = blockIdx.x * blockDim.x + threadIdx.x;
    if (t >= total4) return;
    int base = t * 4;
    const float* vals;
    const int*   idx;
    if (base < src_size) { vals = va; idx = ia + base; }
    else                 { vals = vb; idx = ib + (base - src_size); }
    v4i i4 = __builtin_nontemporal_load(reinterpret_cast<const v4i*>(idx));
    v4f o;
    o.x = vals[i4.x];
    o.y = vals[i4.y];
    o.z = vals[i4.z];
    o.w = vals[i4.w];
    // Regular (RT) store: pass 2 re-reads the table randomly -> keep in L2.
    *reinterpret_cast<v4f*>(table + base) = o;
}

// ---------------------------------------------------------------------------
// Pass 2: out[n] = tanh( sum_k table[concat_idx[8n+k]] * w[k] ), K == 8.
//
// Per wave-iteration: two 16-neuron tiles (32 neurons).
//   lane l -> neuron nl = l&15, K-pair kp = 2*(l>>4)  (0 or 2)
//   A (16x4, rows all == w): VGPR0 = (lane<16 ? w[kp?]..) per documented layout
//       A vgpr0: lanes0-15 K=0, lanes16-31 K=2 ; vgpr1: K=1 / K=3
//   B (4x16): vgpr0 holds rows K=0 (lanes0-15) / K=2 (lanes16-31),
//             vgpr1 holds rows K=1 / K=3  (mirrors documented C/D row layout)
//   Two chained WMMAs accumulate K=0..3 then K=4..7.
//   All D rows identical => d[0] at lane l == out[tile_base + (l&15)].
// ---------------------------------------------------------------------------
__global__ void wmma_rule_kernel(const int*   __restrict__ cix,
                                 const float* __restrict__ table,
                                 const float* __restrict__ w,
                                 float*       __restrict__ out,
                                 int num_pairs) {
    const int  lane = threadIdx.x & 31;
    const int  nl   = lane & 15;
    const int  kp   = (lane >> 4) << 1;   // 0 or 2
    const bool lo   = (lane < 16);

    // Uniform weight reads -> scalar loads.
    const float w0 = w[0], w1 = w[1], w2 = w[2], w3 = w[3];
    const float w4 = w[4], w5 = w[5], w6 = w[6], w7 = w[7];
    v2f aLo, aHi;
    aLo[0] = lo ? w0 : w2;  aLo[1] = lo ? w1 : w3;   // K=0..3
    aHi[0] = lo ? w4 : w6;  aHi[1] = lo ? w5 : w7;   // K=4..7

    const int wave   = (blockIdx.x * blockDim.x + threadIdx.x) >> 5;
    const int nwaves = (gridDim.x * blockDim.x) >> 5;

    for (int p = wave; p < num_pairs; p += nwaves) {
        const int base0 = (p << 5);        // first tile (16 neurons)
        const int base1 = base0 + 16;      // second tile
        const size_t o0 = (size_t)(base0 + nl) * 8 + kp;
        const size_t o1 = (size_t)(base1 + nl) * 8 + kp;

        // Non-temporal: 128 MB index stream must not evict the 16 MB table.
        v2i i0lo = __builtin_nontemporal_load(reinterpret_cast<const v2i*>(cix + o0));
        v2i i0hi = __builtin_nontemporal_load(reinterpret_cast<const v2i*>(cix + o0 + 4));
        v2i i1lo = __builtin_nontemporal_load(reinterpret_cast<const v2i*>(cix + o1));
        v2i i1hi = __builtin_nontemporal_load(reinterpret_cast<const v2i*>(cix + o1 + 4));

        v2f b0lo, b0hi, b1lo, b1hi;        // L2-resident random gathers
        b0lo[0] = table[i0lo.x];  b0lo[1] = table[i0lo.y];
        b0hi[0] = table[i0hi.x];  b0hi[1] = table[i0hi.y];
        b1lo[0] = table[i1lo.x];  b1lo[1] = table[i1lo.y];
        b1hi[0] = table[i1hi.x];  b1hi[1] = table[i1hi.y];

        v8f c0 = {};
        v8f c1 = {};
        c0 = __builtin_amdgcn_wmma_f32_16x16x4_f32(false, aLo, false, b0lo,
                                                   (short)0, c0, false, false);
        c0 = __builtin_amdgcn_wmma_f32_16x16x4_f32(false, aHi, false, b0hi,
                                                   (short)0, c0, false, false);
        c1 = __builtin_amdgcn_wmma_f32_16x16x4_f32(false, aLo, false, b1lo,
                                                   (short)0, c1, false, false);
        c1 = __builtin_amdgcn_wmma_f32_16x16x4_f32(false, aHi, false, b1hi,
                                                   (short)0, c1, false, false);

        const float v    = lo ? c0[0] : c1[0];
        const int   oidx = lo ? (base0 + nl) : (base1 + nl);
        __builtin_nontemporal_store(tanhf(v), out + oidx);
    }
}

// Scalar tail (only used if num_neurons % 32 != 0; not hit for 4M neurons).
__global__ void rule_tail_kernel(const int*   __restrict__ cix,
                                 const float* __restrict__ table,
                                 const float* __restrict__ w,
                                 float*       __restrict__ out,
                                 int start, int n) {
    int i = start + blockIdx.x * blockDim.x + threadIdx.x;
    if (i >= n) return;
    float acc = 0.f;
    for (int k = 0; k < 8; ++k) acc += table[cix[(size_t)i * 8 + k]] * w[k];
    out[i] = tanhf(acc);
}

extern "C" void kernel_launch(void* const* d_in, const int* in_sizes, int n_in,
                              void* d_out, int out_size, void* d_ws, size_t ws_size,
                              hipStream_t stream) {
    const float* va  = (const float*)d_in[0];
    const float* vb  = (const float*)d_in[1];
    const float* w   = (const float*)d_in[2];
    const int*   ia  = (const int*)d_in[3];
    const int*   ib  = (const int*)d_in[4];
    const int*   cix = (const int*)d_in[5];
    float* out   = (float*)d_out;
    float* table = (float*)d_ws;     // 2*SRC_SIZE floats = 16 MB

    const int src     = in_sizes[0];
    const int total   = 2 * src;
    const int neurons = in_sizes[5] / 8;

    // Pass 1: build the concat table.
    const int total4 = total / 4;
    const int tb = 256;
    gather_concat_kernel<<<(total4 + tb - 1) / tb, tb, 0, stream>>>(
        va, vb, ia, ib, table, src, total4);

    // Pass 2: WMMA weighted reduction + tanh. 16384 waves loop over tile pairs
    // (plenty of memory-level parallelism for latency hiding).
    const int pairs = neurons / 32;
    if (pairs > 0) {
        int blocks = 2048;
        if (blocks * 8 > pairs) blocks = (pairs + 7) / 8;  // avoid idle blocks
        if (blocks < 1) blocks = 1;
        wmma_rule_kernel<<<blocks, 256, 0, stream>>>(cix, table, w, out, pairs);
    }
    const int rem_start = pairs * 32;
    if (rem_start < neurons) {
        const int rem = neurons - rem_start;
        rule_tail_kernel<<<(rem + 255) / 256, 256, 0, stream>>>(
            cix, table, w, out, rem_start, neurons);
    }
}